// ODEFunc_cNODE2_DKI_55448027791577
// MI455X (gfx1250) — compile-verified
//
#include <hip/hip_runtime.h>

typedef __attribute__((ext_vector_type(2))) float v2f;
typedef __attribute__((ext_vector_type(8))) float v8f;

#define BM 128
#define BN 128
#define BK 32
#define PITCH 36   // even + 16B-aligned rows for B128 async writes; 36*i mod 64 distinct banks

__device__ __forceinline__ void wait_asynccnt0() {
#if __has_builtin(__builtin_amdgcn_s_wait_asynccnt)
  __builtin_amdgcn_s_wait_asynccnt(0);
#else
  asm volatile("s_wait_asynccnt 0x0" ::: "memory");
#endif
}

// One b128 async DMA: 16 bytes/lane from the global address base + goff to the LDS address in ldsoff.
__device__ __forceinline__ void async_b128(unsigned ldsoff, unsigned goff,
                                           const float* base) {
  asm volatile("global_load_async_to_lds_b128 %0, %1, %2"
               :: "v"(ldsoff), "v"(goff), "s"(base)
               : "memory");
}

// C[M,N] = A[M,K] * W[N,K]^T + bias[N]
// 256 threads = 8 waves, wave grid 4(M) x 2(N); each wave: 32x64 tile
// = 2x4 WMMA 16x16 f32 accumulators, K stepped by 4 (V_WMMA_F32_16X16X4_F32).
// Double-buffered LDS staged via GLOBAL_LOAD_ASYNC_TO_LDS_B128 (ASYNCcnt).
__global__ __launch_bounds__(256)
void gemm_bias_kernel(const float* __restrict__ A, const float* __restrict__ W,
                      const float* __restrict__ bias, float* __restrict__ C,
                      int M, int N, int K) {
  __shared__ float As[2][BM * PITCH];
  __shared__ float Ws[2][BN * PITCH];

  const int tid   = threadIdx.x;
  const int wave  = tid >> 5;
  const int lane  = tid & 31;
  const int waveM = wave >> 1;   // 0..3
  const int waveN = wave & 1;    // 0..1

  const int blockM = blockIdx.x * BM;
  const int blockN = blockIdx.y * BN;

  const int l16   = lane & 15;
  const int hi    = lane >> 4;   // 0 or 1
  const int khalf = hi * 2;      // K sub-offset for 16x4 frag layout

  // Per-thread DMA coordinates: 128 rows x 8 float4 = 1024 transfers, 4/thread.
  unsigned goA[4], goW[4], ldsOff[4];
#pragma unroll
  for (int i = 0; i < 4; ++i) {
    const int f   = tid + i * 256;
    const int row = f >> 3;
    const int c4  = (f & 7) * 4;
    goA[i]    = (unsigned)(((size_t)(blockM + row) * K + c4) * 4);
    goW[i]    = (unsigned)(((size_t)(blockN + row) * K + c4) * 4);
    ldsOff[i] = (unsigned)(row * PITCH + c4) * 4u;
  }
  const unsigned ldsA0 = (unsigned)(size_t)&As[0][0];
  const unsigned ldsW0 = (unsigned)(size_t)&Ws[0][0];
  const unsigned bufStride = (unsigned)(BM * PITCH * 4);

  v8f acc[2][4];
#pragma unroll
  for (int mt = 0; mt < 2; ++mt)
#pragma unroll
    for (int nt = 0; nt < 4; ++nt)
      acc[mt][nt] = (v8f){0.f, 0.f, 0.f, 0.f, 0.f, 0.f, 0.f, 0.f};

  auto issue_tile = [&](int buf, int k0) {
    const unsigned kb = (unsigned)(k0 * 4);
    const unsigned bs = (unsigned)buf * bufStride;
#pragma unroll
    for (int i = 0; i < 4; ++i)
      async_b128(ldsA0 + bs + ldsOff[i], goA[i] + kb, A);
#pragma unroll
    for (int i = 0; i < 4; ++i)
      async_b128(ldsW0 + bs + ldsOff[i], goW[i] + kb, W);
  };

  const int NT = K / BK;
  issue_tile(0, 0);

  for (int kt = 0; kt < NT; ++kt) {
    const int cur = kt & 1;
    wait_asynccnt0();     // this wave's DMA into buffer `cur` done
    __syncthreads();      // everyone's DMA done + prev compute on `cur^1` done
    if (kt + 1 < NT) issue_tile(cur ^ 1, (kt + 1) * BK);

    const float* __restrict__ as = As[cur];
    const float* __restrict__ ws = Ws[cur];

#pragma unroll
    for (int ks = 0; ks < BK; ks += 4) {
      v2f afrag[2];
#pragma unroll
      for (int mt = 0; mt < 2; ++mt) {
        const int m = waveM * 32 + mt * 16 + l16;
        afrag[mt].x = as[m * PITCH + ks + khalf];
        afrag[mt].y = as[m * PITCH + ks + khalf + 1];
      }
      v2f bfrag[4];
#pragma unroll
      for (int nt = 0; nt < 4; ++nt) {
        const int n = waveN * 64 + nt * 16 + l16;
        bfrag[nt].x = ws[n * PITCH + ks + khalf];
        bfrag[nt].y = ws[n * PITCH + ks + khalf + 1];
      }
#pragma unroll
      for (int mt = 0; mt < 2; ++mt)
#pragma unroll
        for (int nt = 0; nt < 4; ++nt)
          acc[mt][nt] = __builtin_amdgcn_wmma_f32_16x16x4_f32(
              false, afrag[mt], false, bfrag[nt],
              (short)0, acc[mt][nt], false, false);
    }
  }

  // Writeback with bias. C layout: lane holds N=l16, VGPR v holds M=v+8*hi.
#pragma unroll
  for (int mt = 0; mt < 2; ++mt) {
#pragma unroll
    for (int nt = 0; nt < 4; ++nt) {
      const int gn = blockN + waveN * 64 + nt * 16 + l16;
      const float bv = bias[gn];
#pragma unroll
      for (int v = 0; v < 8; ++v) {
        const int gm = blockM + waveM * 32 + mt * 16 + v + 8 * hi;
        C[(size_t)gm * N + gn] = acc[mt][nt][v] + bv;
      }
    }
  }
}

// dot[row] = sum_n y[row][n] * out[row][n]
__global__ __launch_bounds__(256)
void row_dot_kernel(const float* __restrict__ y, const float* __restrict__ out,
                    float* __restrict__ dot, int N) {
  __shared__ float red[256];
  const int row = blockIdx.x;
  const float4* yr = (const float4*)(y + (size_t)row * N);
  const float4* orow = (const float4*)(out + (size_t)row * N);
  float s = 0.f;
  for (int i = threadIdx.x; i < (N >> 2); i += 256) {
    const float4 a = yr[i];
    const float4 b = orow[i];
    s += a.x * b.x + a.y * b.y + a.z * b.z + a.w * b.w;
  }
  red[threadIdx.x] = s;
  __syncthreads();
  for (int off = 128; off > 0; off >>= 1) {
    if (threadIdx.x < off) red[threadIdx.x] += red[threadIdx.x + off];
    __syncthreads();
  }
  if (threadIdx.x == 0) dot[row] = red[0];
}

// out[b][n] = y[b][n] * (out[b][n] - dot[b]),  processed as float4
__global__ __launch_bounds__(256)
void final_kernel(const float* __restrict__ y, const float* __restrict__ dot,
                  float* __restrict__ out, int N) {
  const size_t idx = (size_t)blockIdx.x * 256 + threadIdx.x;  // float4 index
  const int row = (int)((idx * 4) / (size_t)N);
  const float d = dot[row];
  const float4 yv = ((const float4*)y)[idx];
  float4 ov = ((float4*)out)[idx];
  ov.x = yv.x * (ov.x - d);
  ov.y = yv.y * (ov.y - d);
  ov.z = yv.z * (ov.z - d);
  ov.w = yv.w * (ov.w - d);
  ((float4*)out)[idx] = ov;
}

extern "C" void kernel_launch(void* const* d_in, const int* in_sizes, int n_in,
                              void* d_out, int out_size, void* d_ws, size_t ws_size,
                              hipStream_t stream) {
  // inputs: t, y, W1, b1, W2, b2
  const float* y  = (const float*)d_in[1];
  const float* W1 = (const float*)d_in[2];
  const float* b1 = (const float*)d_in[3];
  const float* W2 = (const float*)d_in[4];
  const float* b2 = (const float*)d_in[5];
  float* out = (float*)d_out;

  const int N = in_sizes[3];              // 4096
  const int B = in_sizes[1] / N;          // 2048

  float* h   = (float*)d_ws;                       // B*N floats
  float* dot = (float*)d_ws + (size_t)B * N;       // B floats

  dim3 blk(256);
  dim3 grid1(B / BM, N / BN);

  // h = y @ W1^T + b1
  gemm_bias_kernel<<<grid1, blk, 0, stream>>>(y, W1, b1, h, B, N, N);
  // out = h @ W2^T + b2
  gemm_bias_kernel<<<grid1, blk, 0, stream>>>(h, W2, b2, out, B, N, N);
  // dot[b] = <y_b, out_b>
  row_dot_kernel<<<dim3(B), blk, 0, stream>>>(y, out, dot, N);
  // out = y * (out - dot)
  const size_t nv4 = ((size_t)B * N) / 4;
  final_kernel<<<dim3((unsigned)(nv4 / 256)), blk, 0, stream>>>(y, dot, out, N);
}